// Matcher_74955769250598
// MI455X (gfx1250) — compile-verified
//
#include <hip/hip_runtime.h>
#include <math.h>

typedef float v2f __attribute__((ext_vector_type(2)));
typedef float v8f __attribute__((ext_vector_type(8)));

#define B_      32
#define L_      128
#define D_      256
#define T_      10000
#define NSTRIP  625      // T_/16
#define WPB     79       // ceil(NSTRIP/8) workgroups per batch row
#define KCH     64       // K columns staged in LDS per stage
#define SP      68       // padded LDS row stride (floats): 68 % 64 == 4 -> conflict-free b64 reads
#define NSTAGE  4        // D_/KCH

#if defined(__has_builtin)
#if __has_builtin(__builtin_amdgcn_global_load_async_to_lds_b128)
#define HAVE_ASYNC_LDS 1
#else
#define HAVE_ASYNC_LDS 0
#endif
#else
#define HAVE_ASYNC_LDS 0
#endif

// 16-byte int vector matching the builtin's parameter type exactly
typedef int vs4i __attribute__((vector_size(16)));
typedef __attribute__((address_space(1))) vs4i gv4i;   // global
typedef __attribute__((address_space(3))) vs4i lv4i;   // LDS

__device__ __forceinline__ void wait_async_all() {
#if HAVE_ASYNC_LDS
#if __has_builtin(__builtin_amdgcn_s_wait_asynccnt)
    __builtin_amdgcn_s_wait_asynccnt(0);
#else
    asm volatile("s_wait_asynccnt 0x0" ::: "memory");
#endif
#endif
}

// Issue one K-stage of the gathered embedding tile into an LDS buffer.
// 128 rows x 64 cols staged at float4/b128 granularity (2048 vec4 elements).
__device__ __forceinline__ void stage_issue(const float* __restrict__ emb,
                                            const int* sev, float* buf,
                                            int st, int tid) {
    for (int i = tid; i < L_ * (KCH / 4); i += 256) {
        int l  = i >> 4;                 // /(KCH/4)
        int c4 = (i & 15) << 2;
        const float* src = emb + (size_t)sev[l] * D_ + st * KCH + c4;
        float* dst = buf + l * SP + c4;
#if HAVE_ASYNC_LDS
        __builtin_amdgcn_global_load_async_to_lds_b128(
            (gv4i*)src, (lv4i*)dst, 0, 0);
#else
        *(float4*)dst = *(const float4*)src;
#endif
    }
}

// ---------------- kernel 1: dm[b,d] = mean_l data[b,l,d] ----------------
__global__ __launch_bounds__(256) void k_dm(const float* __restrict__ data,
                                            float* __restrict__ dm) {
    int b = blockIdx.x, d = threadIdx.x;                 // blockDim.x == D_
    const float* p = data + (size_t)b * L_ * D_ + d;
    float s = 0.f;
    for (int l = 0; l < L_; ++l) s += p[(size_t)l * D_];
    dm[b * D_ + d] = s * (1.0f / L_);
}

// ---- kernel 2: fused  tanh(enc·W5^T) * drop * corr, reduced over L ----
// grid: B_*WPB blocks of 256 (8 waves). Each wave owns a 16-wide t strip.
// Double-buffered async LDS staging of gathered embeddings.
__global__ __launch_bounds__(256) void k_item(const float* __restrict__ emb,
                                              const int*   __restrict__ etype,
                                              const float* __restrict__ W5,
                                              const float* __restrict__ corr,
                                              const float* __restrict__ drop,
                                              float* __restrict__ item_pre) {
    __shared__ float As[2][L_ * SP];
    __shared__ int   sev[L_];

    const int b    = blockIdx.x / WPB;
    const int sblk = blockIdx.x % WPB;
    const int tid  = threadIdx.x;
    const int wid  = tid >> 5;
    const int lane = tid & 31;
    const int strip = sblk * 8 + wid;
    const bool active = (strip < NSTRIP);
    const int t0 = strip * 16;
    const int lr = lane & 15;       // N / M row within fragment
    const int lh = lane >> 4;       // lane half: K pair (A/B) or +8 rows (C)

    if (tid < L_) sev[tid] = etype[b * L_ + tid];
    __syncthreads();                          // sev visible to stagers

    stage_issue(emb, sev, As[0], 0, tid);     // prologue: stage 0
    wait_async_all();
    __syncthreads();

    v8f acc[8] = {};                // 8 M-tiles (l = 0..127) x 16 t columns

    for (int st = 0; st < NSTAGE; ++st) {
        const int cur = st & 1;
        if (st + 1 < NSTAGE)
            stage_issue(emb, sev, As[cur ^ 1], st + 1, tid);  // prefetch next
        if (active) {
            const float* __restrict__ Ab = As[cur];
            for (int ks = 0; ks < KCH / 4; ++ks) {
                const int kk = ks * 4 + 2 * lh;          // local k for this lane
                const int kg = st * KCH + kk;            // global k
                // B fragment: B[k][n] = W5[t0+n][k]
                v2f bf = *(const v2f*)(W5 + (size_t)(t0 + lr) * D_ + kg);
                #pragma unroll
                for (int m = 0; m < 8; ++m) {
                    v2f af = *(const v2f*)(&Ab[(m * 16 + lr) * SP + kk]);
                    acc[m] = __builtin_amdgcn_wmma_f32_16x16x4_f32(
                        false, af, false, bf, (short)0, acc[m], false, false);
                }
            }
        }
        wait_async_all();           // next-stage data landed in LDS
        __syncthreads();
    }

    if (active) {
        const int t = t0 + lr;
        float s = 0.f;
        #pragma unroll
        for (int m = 0; m < 8; ++m) {
            #pragma unroll
            for (int r = 0; r < 8; ++r) {
                int l  = m * 16 + r + lh * 8;            // C layout: VGPR r -> M=r(+8)
                int ev = sev[l];
                float w = 0.f;
                if (ev != 0) {
                    float dv = __builtin_nontemporal_load(
                        &drop[(size_t)(b * L_ + l) * T_ + t]);     // streamed once
                    w = dv * corr[(size_t)(ev - 1) * T_ + t];      // L2-resident
                }
                s += tanhf(acc[m][r]) * w;
            }
        }
        s += __shfl_xor(s, 16, 32);                      // merge l halves
        if (lh == 0) item_pre[(size_t)b * T_ + t] = s;
    }
}

// -------- kernel 3: q[w][b][t] = dm · W_w^T  (four GEMMs, WMMA) --------
__global__ __launch_bounds__(256) void k_pw(const float* __restrict__ dm,
                                            const float* __restrict__ W1,
                                            const float* __restrict__ W2,
                                            const float* __restrict__ W3,
                                            const float* __restrict__ W4,
                                            float* __restrict__ q) {
    const int tid = threadIdx.x, wid = tid >> 5, lane = tid & 31;
    const int strip = blockIdx.x * 8 + wid;
    if (strip >= NSTRIP) return;
    const int t0 = strip * 16;
    const int lr = lane & 15, lh = lane >> 4;
    const float* Ws[4] = {W1, W2, W3, W4};

    v8f acc[4][2] = {};
    for (int ks = 0; ks < D_ / 4; ++ks) {
        const int kk = ks * 4 + 2 * lh;
        v2f af0 = *(const v2f*)(dm + (size_t)(lr) * D_ + kk);        // rows 0..15
        v2f af1 = *(const v2f*)(dm + (size_t)(16 + lr) * D_ + kk);   // rows 16..31
        #pragma unroll
        for (int w = 0; w < 4; ++w) {
            v2f bf = *(const v2f*)(Ws[w] + (size_t)(t0 + lr) * D_ + kk);
            acc[w][0] = __builtin_amdgcn_wmma_f32_16x16x4_f32(
                false, af0, false, bf, (short)0, acc[w][0], false, false);
            acc[w][1] = __builtin_amdgcn_wmma_f32_16x16x4_f32(
                false, af1, false, bf, (short)0, acc[w][1], false, false);
        }
    }
    const int t = t0 + lr;
    #pragma unroll
    for (int w = 0; w < 4; ++w)
        #pragma unroll
        for (int m = 0; m < 2; ++m)
            #pragma unroll
            for (int r = 0; r < 8; ++r) {
                int brow = m * 16 + r + lh * 8;
                q[((size_t)w * B_ + brow) * T_ + t] = acc[w][m][r];
            }
}

// ---- kernel 4: row l2-norms (5x) + weighted tanh combine; target=1 ----
__global__ __launch_bounds__(256) void k_combine(const float* __restrict__ item_pre,
                                                 const float* __restrict__ q,
                                                 const float* __restrict__ pa,
                                                 const float* __restrict__ pb,
                                                 const float* __restrict__ pc,
                                                 const float* __restrict__ pd,
                                                 float* __restrict__ out,
                                                 float* __restrict__ target) {
    const int b = blockIdx.x, tid = threadIdx.x;
    const float* qi = item_pre + (size_t)b * T_;
    const float* q1 = q + ((size_t)0 * B_ + b) * T_;
    const float* q2 = q + ((size_t)1 * B_ + b) * T_;
    const float* q3 = q + ((size_t)2 * B_ + b) * T_;
    const float* q4 = q + ((size_t)3 * B_ + b) * T_;

    float s0 = 0.f, s1 = 0.f, s2 = 0.f, s3 = 0.f, s4 = 0.f;
    for (int t = tid; t < T_; t += 256) {
        float v;
        v = qi[t]; s0 += v * v;
        v = q1[t]; s1 += v * v;
        v = q2[t]; s2 += v * v;
        v = q3[t]; s3 += v * v;
        v = q4[t]; s4 += v * v;
    }
    __shared__ float red[5][256];
    red[0][tid] = s0; red[1][tid] = s1; red[2][tid] = s2;
    red[3][tid] = s3; red[4][tid] = s4;
    __syncthreads();
    for (int off = 128; off > 0; off >>= 1) {
        if (tid < off)
            #pragma unroll
            for (int j = 0; j < 5; ++j) red[j][tid] += red[j][tid + off];
        __syncthreads();
    }
    const float ni = 1.f / fmaxf(sqrtf(red[0][0]), 1e-5f);
    const float n1 = 1.f / fmaxf(sqrtf(red[1][0]), 1e-5f);
    const float n2 = 1.f / fmaxf(sqrtf(red[2][0]), 1e-5f);
    const float n3 = 1.f / fmaxf(sqrtf(red[3][0]), 1e-5f);
    const float n4 = 1.f / fmaxf(sqrtf(red[4][0]), 1e-5f);
    const float ca = pa[0], cb = pb[0], cc = pc[0], cd = pd[0];
    for (int t = tid; t < T_; t += 256) {
        float v = q1[t] * n1 * ca + q2[t] * n2 * cb + q3[t] * n3 * cc +
                  q4[t] * n4 * cd + qi[t] * ni;
        __builtin_nontemporal_store(tanhf(v), &out[(size_t)b * T_ + t]);
        __builtin_nontemporal_store(1.0f,     &target[(size_t)b * T_ + t]);
    }
}

// -------- kernel 5: scatter zeros into target at observed events --------
__global__ __launch_bounds__(256) void k_target(const int* __restrict__ etype,
                                                float* __restrict__ target) {
    int i = blockIdx.x * blockDim.x + threadIdx.x;
    if (i >= B_ * L_) return;
    int ev = etype[i];
    if (ev != 0) target[(size_t)(i / L_) * T_ + (ev - 1)] = 0.0f;
}

extern "C" void kernel_launch(void* const* d_in, const int* in_sizes, int n_in,
                              void* d_out, int out_size, void* d_ws, size_t ws_size,
                              hipStream_t stream) {
    const float* data  = (const float*)d_in[0];
    const int*   etype = (const int*)  d_in[1];
    const float* emb   = (const float*)d_in[2];
    const float* corr  = (const float*)d_in[3];
    const float* W1    = (const float*)d_in[4];
    const float* W2    = (const float*)d_in[5];
    const float* W3    = (const float*)d_in[6];
    const float* W4    = (const float*)d_in[7];
    const float* W5    = (const float*)d_in[8];
    const float* pa    = (const float*)d_in[9];
    const float* pb    = (const float*)d_in[10];
    const float* pc    = (const float*)d_in[11];
    const float* pd    = (const float*)d_in[12];
    const float* drop  = (const float*)d_in[13];

    float* out    = (float*)d_out;            // [B,T]
    float* target = out + (size_t)B_ * T_;    // [B,T]

    float* dm       = (float*)d_ws;           // B_*D_
    float* item_pre = dm + B_ * D_;           // B_*T_
    float* q        = item_pre + B_ * T_;     // 4*B_*T_

    k_dm     <<<B_,        256, 0, stream>>>(data, dm);
    k_item   <<<B_ * WPB,  256, 0, stream>>>(emb, etype, W5, corr, drop, item_pre);
    k_pw     <<<WPB,       256, 0, stream>>>(dm, W1, W2, W3, W4, q);
    k_combine<<<B_,        256, 0, stream>>>(item_pre, q, pa, pb, pc, pd, out, target);
    k_target <<<(B_ * L_ + 255) / 256, 256, 0, stream>>>(etype, target);
}